// ModularHardVoting_86998857547973
// MI455X (gfx1250) — compile-verified
//
#include <hip/hip_runtime.h>

// Problem constants from the reference.
#define BATCH 256
#define SEQ   1024
#define EMB   256
#define NCLS  4
#define NBR   4

typedef __attribute__((ext_vector_type(2))) float v2f;
typedef __attribute__((ext_vector_type(4))) float v4f;
typedef __attribute__((ext_vector_type(8))) float v8f;

// ---------------------------------------------------------------------------
// Kernel 1: streaming mean over the sequence dimension.
// One block per (batch, branch) pair. 256 threads:
//   thread t: e4 = t & 63  -> which float4 column chunk of E
//             sg = t >> 6  -> which of 4 interleaved S row-groups
// Every iteration the block loads 4 rows x 256 floats = 4KB, coalesced B128.
// Non-temporal: the 1GB stream is touched exactly once (>> 192MB L2).
// ---------------------------------------------------------------------------
__global__ void __launch_bounds__(256)
pool_mean_kernel(const float* __restrict__ x0, const float* __restrict__ x1,
                 const float* __restrict__ x2, const float* __restrict__ x3,
                 float* __restrict__ pooled) {
    const int bb = blockIdx.x;        // 0..BATCH*NBR-1, bb = b*NBR + n
    const int b  = bb >> 2;
    const int n  = bb & 3;
    const float* xb = (n == 0) ? x0 : (n == 1) ? x1 : (n == 2) ? x2 : x3;
    const v4f* base = (const v4f*)(xb + (size_t)b * SEQ * EMB);

    const int t  = threadIdx.x;
    const int e4 = t & 63;            // float4 chunk in E (E/4 = 64)
    const int sg = t >> 6;            // 0..3

    v4f acc = {0.f, 0.f, 0.f, 0.f};
    const v4f* p = base + e4;
    #pragma unroll 4
    for (int s = sg; s < SEQ; s += 4) {
        v4f v = __builtin_nontemporal_load(p + (size_t)s * (EMB / 4));
        acc += v;
    }

    __shared__ v4f red[4][64];
    red[sg][e4] = acc;
    __syncthreads();
    if (sg == 0) {
        v4f s = red[0][e4] + red[1][e4] + red[2][e4] + red[3][e4];
        s *= (1.0f / (float)SEQ);
        ((v4f*)pooled)[(size_t)bb * (EMB / 4) + e4] = s;
    }
}

// ---------------------------------------------------------------------------
// Kernel 2: pooled[1024x256] @ W^T[256x4] via V_WMMA_F32_16X16X4_F32,
// then bias + ReLU + argmax (softmax is monotone) + hard-vote combine.
// One wave of 32 lanes handles 16 rows (= 4 batches x 4 branches).
// A-matrix (16x4 f32, 2 VGPRs): lanes 0-15 -> M=lane, K={0,1};
//                               lanes 16-31 -> M=lane-16, K={2,3}.
// B-matrix (4x16 f32, 2 VGPRs): mirrored; columns >= NCLS zeroed via
// v_cndmask on the loaded value (unconditional clamped load -> no EXEC
// branches in the WMMA loop; WMMA requires EXEC all-1s anyway).
// D (8 VGPRs): VGPR j holds M=j (lanes 0-15) / M=8+j (lanes 16-31), N=lane%16.
// ---------------------------------------------------------------------------
__global__ void __launch_bounds__(128)
linear_vote_kernel(const float* __restrict__ pooled, const float* __restrict__ W,
                   const float* __restrict__ bias, float* __restrict__ out) {
    const int lane = threadIdx.x & 31;
    const int wib  = threadIdx.x >> 5;           // wave in block (0..3)
    const int wave = blockIdx.x * 4 + wib;       // 0..63
    const int row0 = wave * 16;                  // first pooled row of tile
    const int m    = lane & 15;
    const int kh   = (lane >> 4) << 1;           // 0 for lanes 0-15, 2 for 16-31

    const float* arow = pooled + (size_t)(row0 + m) * EMB;
    const bool   bval = (m < NCLS);
    const int    mcl  = bval ? m : (NCLS - 1);   // clamped: always in-bounds
    const float* wrow = W + (size_t)mcl * EMB;   // W is [NCLS][EMB] row-major

    v8f acc = {};
    for (int kk = 0; kk < EMB; kk += 4) {
        v2f a  = *(const v2f*)(arow + kk + kh);  // global_load_b64
        v2f wv = *(const v2f*)(wrow + kk + kh);  // global_load_b64 (clamped)
        v2f bm;
        bm.x = bval ? wv.x : 0.0f;               // v_cndmask, no EXEC branch
        bm.y = bval ? wv.y : 0.0f;
        acc = __builtin_amdgcn_wmma_f32_16x16x4_f32(
            /*neg_a=*/false, a, /*neg_b=*/false, bm,
            /*c_mod=*/(short)0, acc, /*reuse_a=*/false, /*reuse_b=*/false);
    }

    // Spill D columns 0..3 to LDS: lg[wave][row][class]
    __shared__ float lg[4][16][NCLS];
    const int nd  = lane & 15;
    const int mhi = (lane >> 4) << 3;
    if (nd < NCLS) {
        #pragma unroll
        for (int j = 0; j < 8; ++j)
            lg[wib][mhi + j][nd] = acc[j];
    }
    __syncthreads();

    // Per-row finalize: bias + ReLU + first-max argmax (matches jnp.argmax
    // tie-breaking, including the all-ReLU-zero case).
    int pred = 0;
    if (lane < 16) {
        float best = -1.0f;
        #pragma unroll
        for (int c = 0; c < NCLS; ++c) {
            float v = lg[wib][lane][c] + bias[c];
            v = fmaxf(v, 0.0f);
            if (v > best) { best = v; pred = c; }
        }
    }

    // Rows row0+lane: row = b*NBR + n, so lanes {4k..4k+3} share one batch.
    const int base4 = lane & ~3;
    const int p0 = __shfl(pred, base4 + 0, 32);
    const int p1 = __shfl(pred, base4 + 1, 32);
    const int p2 = __shfl(pred, base4 + 2, 32);
    const int p3 = __shfl(pred, base4 + 3, 32);

    if (lane < 16 && (lane & 3) == 0) {
        v4f o;
        #pragma unroll
        for (int c = 0; c < NCLS; ++c) {
            int cnt = (p0 == c) + (p1 == c) + (p2 == c) + (p3 == c);
            o[c] = (float)cnt * 0.25f;
        }
        const int bidx = (row0 + lane) >> 2;     // batch index
        ((v4f*)out)[bidx] = o;                    // writes out[bidx*4 .. +3]
    }
}

extern "C" void kernel_launch(void* const* d_in, const int* in_sizes, int n_in,
                              void* d_out, int out_size, void* d_ws, size_t ws_size,
                              hipStream_t stream) {
    (void)in_sizes; (void)n_in; (void)out_size; (void)ws_size;
    const float* x_acc  = (const float*)d_in[0];
    const float* x_bvp  = (const float*)d_in[1];
    const float* x_eda  = (const float*)d_in[2];
    const float* x_temp = (const float*)d_in[3];
    const float* W      = (const float*)d_in[4];
    const float* bias   = (const float*)d_in[5];
    float* out    = (float*)d_out;
    float* pooled = (float*)d_ws;   // 1024 x 256 f32 = 1 MB scratch

    // Phase 1: bandwidth-bound mean over S (reads 1.07 GB once).
    pool_mean_kernel<<<BATCH * NBR, 256, 0, stream>>>(x_acc, x_bvp, x_eda, x_temp, pooled);

    // Phase 2: WMMA linear + ReLU + argmax + hard voting (tiny).
    linear_vote_kernel<<<(BATCH * NBR) / (16 * 4), 128, 0, stream>>>(pooled, W, bias, out);
}